// Subtraction_40475771798075
// MI455X (gfx1250) — compile-verified
//
#include <hip/hip_runtime.h>
#include <cstdint>
#include <cstddef>

// Problem constants (match reference)
#define N_      4
#define C_      64
#define H_      112
#define W_      112
#define Kk      7
#define PADr    3
#define HW      (H_ * W_)        // 12544 output pixels per plane
#define KK2     (Kk * Kk)        // 49 neighbor offsets
#define ROWS    16               // output rows per block
#define RB      (H_ / ROWS)      // 7 row-blocks per plane
#define TILE_H  (ROWS + 2 * PADr)   // 22
#define TILE_W  (W_ + 2 * PADr)     // 118
#define LDW     120              // padded LDS row pitch (floats)
#define TELEMS  (TILE_H * TILE_W)   // 2596 tile elements
#define NTHREADS 256

// Native clang vector type: accepted by __builtin_nontemporal_store,
// lowers to one global_store_b128.
typedef float v4f __attribute__((ext_vector_type(4)));

__device__ __forceinline__ int reflect_idx(int t) {
    // reflect padding, valid for t in [-PADr, H_+PADr)
    t = (t < 0) ? -t : t;
    t = (t >= H_) ? (2 * H_ - 2 - t) : t;
    return t;
}

__global__ __launch_bounds__(NTHREADS)
void san_subtraction_kernel(const float* __restrict__ x, float* __restrict__ out) {
    __shared__ float tile[TILE_H * LDW];   // 10,560 bytes

    const int bx    = blockIdx.x;
    const int plane = bx / RB;            // n*C + c  (0..255)
    const int rb    = bx - plane * RB;
    const int row0  = rb * ROWS;

    const float* __restrict__ src = x + (size_t)plane * HW;

    // ---------------------------------------------------------------
    // Phase 1: async gather of the reflect-padded tile into LDS.
    // Per-lane global address (reflect-mapped) -> per-lane LDS address.
    // Tracked by ASYNCcnt; no VGPR staging of the data.
    // ---------------------------------------------------------------
    for (int e = threadIdx.x; e < TELEMS; e += NTHREADS) {
        const int ph = e / TILE_W;
        const int pw = e - ph * TILE_W;
        const int h  = reflect_idx(row0 + ph - PADr);
        const int w  = reflect_idx(pw - PADr);
        const float* g = src + h * W_ + w;
        // Low 32 bits of the flat shared-aperture address == LDS byte offset.
        unsigned lds_off =
            (unsigned)(unsigned long long)(uintptr_t)(&tile[ph * LDW + pw]);
        asm volatile("global_load_async_to_lds_b32 %0, %1, off"
                     :: "v"(lds_off), "v"(g)
                     : "memory");
    }
    asm volatile("s_wait_asynccnt 0x0" ::: "memory");
    __syncthreads();

    // ---------------------------------------------------------------
    // Phase 2: for each float4 of output pixels, emit 49 differences.
    // Stores are write-once streaming (629 MB >> 192 MB L2) -> non-temporal
    // 128-bit stores, coalesced across the wave.
    // ---------------------------------------------------------------
    const int W4    = W_ / 4;        // 28 quads per row
    const int QUADS = ROWS * W4;     // 448 quads per block
    float* __restrict__ obase = out + (size_t)plane * KK2 * HW;

    for (int q = threadIdx.x; q < QUADS; q += NTHREADS) {
        const int hl = q / W4;
        const int wl = (q - hl * W4) * 4;

        const float* crow = &tile[(hl + PADr) * LDW + (wl + PADr)];
        const float c0 = crow[0];
        const float c1 = crow[1];
        const float c2 = crow[2];
        const float c3 = crow[3];

        const int p = (row0 + hl) * W_ + wl;

        #pragma unroll
        for (int kh = 0; kh < Kk; ++kh) {
            #pragma unroll
            for (int kw = 0; kw < Kk; ++kw) {
                const float* trow = &tile[(hl + kh) * LDW + (wl + kw)];
                v4f v;
                v.x = c0 - trow[0];
                v.y = c1 - trow[1];
                v.z = c2 - trow[2];
                v.w = c3 - trow[3];
                const int kk = kh * Kk + kw;
                v4f* dst = (v4f*)(obase + (size_t)kk * HW + p);
                __builtin_nontemporal_store(v, dst);
            }
        }
    }
}

extern "C" void kernel_launch(void* const* d_in, const int* in_sizes, int n_in,
                              void* d_out, int out_size, void* d_ws, size_t ws_size,
                              hipStream_t stream) {
    (void)in_sizes; (void)n_in; (void)d_ws; (void)ws_size; (void)out_size;
    const float* x  = (const float*)d_in[0];
    float*      out = (float*)d_out;

    dim3 grid(N_ * C_ * RB);   // 1792 blocks: one per (plane, 16-row band)
    dim3 block(NTHREADS);      // 8 wave32 waves
    san_subtraction_kernel<<<grid, block, 0, stream>>>(x, out);
}